// EmbeddingManager_19439021981897
// MI455X (gfx1250) — compile-verified
//
#include <hip/hip_runtime.h>
#include <hip/hip_bf16.h>

// Masked row-select: out[b,n,:] = (tok[b,n]==ph_tok) ? ph_emb[b,:] : emb[b,n,:]
// Pure bandwidth kernel: ~970 MB total traffic -> ~42us floor at 23.3 TB/s.
// Strategy: 192 threads/block (6 wave32 waves) = one 768-float row in float4s,
// 4 rows per block, wave-uniform mask (scalarized token load + SALU branch, no
// lane divergence), NT hints on the two streaming tensors so the reusable 6MB
// placeholder table stays resident in the 192MB L2.

typedef float v4f __attribute__((ext_vector_type(4)));

namespace {
constexpr int kN          = 77;   // tokens per row
constexpr int kD4         = 192;  // 768 floats / 4 per float4
constexpr int kRowsPerBlk = 4;
}

__device__ __forceinline__ void do_row(long long row, int t, int ph_tok,
                                       const int* __restrict__ tok,
                                       const v4f* __restrict__ emb,
                                       const v4f* __restrict__ ph_emb,
                                       v4f* __restrict__ out)
{
    const int  b     = (int)(row / kN);          // batch index (uniform)
    const bool is_ph = (tok[row] == ph_tok);     // uniform scalar load -> SALU branch

    v4f v;
    if (is_ph) {
        // Reused 77x per batch row; 6MB total -> regular temporal (L2 resident).
        v = ph_emb[(long long)b * kD4 + t];
    } else {
        // Read-once stream: non-temporal, don't thrash L2.
        v = __builtin_nontemporal_load(&emb[row * kD4 + t]);
    }
    // Write-once stream: non-temporal store.
    __builtin_nontemporal_store(v, &out[row * kD4 + t]);
}

__global__ __launch_bounds__(192)
void EmbeddingManager_19439021981897_kernel(const int*  __restrict__ tok,      // [B*N]
                                            const v4f*  __restrict__ emb,      // [B*N, 192]
                                            const v4f*  __restrict__ ph_emb,   // [B, 192]
                                            const int*  __restrict__ ph_tok_p, // [1]
                                            v4f*        __restrict__ out,      // [B*N, 192]
                                            int total_rows)
{
    const int t = threadIdx.x;                                   // 0..191, float4 slot in row
    const long long row0 = (long long)blockIdx.x * kRowsPerBlk;
    const int ph_tok = *ph_tok_p;                                // uniform scalar load

    if (row0 + kRowsPerBlk <= total_rows) {
        // Hot path: full block, no per-row bounds checks.
#pragma unroll
        for (int r = 0; r < kRowsPerBlk; ++r)
            do_row(row0 + r, t, ph_tok, tok, emb, ph_emb, out);
    } else {
        // Tail block (doesn't occur for B*N=157696, kept for safety).
        for (int r = 0; r < kRowsPerBlk; ++r) {
            const long long row = row0 + r;
            if (row >= total_rows) return;
            do_row(row, t, ph_tok, tok, emb, ph_emb, out);
        }
    }
}

extern "C" void kernel_launch(void* const* d_in, const int* in_sizes, int n_in,
                              void* d_out, int out_size, void* d_ws, size_t ws_size,
                              hipStream_t stream) {
    (void)n_in; (void)d_ws; (void)ws_size; (void)out_size;

    const int* tok    = (const int*)d_in[0];   // tokenized_text  [B*N]
    const v4f* emb    = (const v4f*)d_in[1];   // embedded_text   [B*N*768] fp32
    const v4f* ph_emb = (const v4f*)d_in[2];   // placeholder_emb [B*768]   fp32
    const int* ph_tok = (const int*)d_in[3];   // placeholder_token scalar
    v4f*       out    = (v4f*)d_out;

    const int total_rows = in_sizes[0];        // B*N = 157696
    const int blocks     = (total_rows + kRowsPerBlk - 1) / kRowsPerBlk;  // 39424

    EmbeddingManager_19439021981897_kernel<<<blocks, 192, 0, stream>>>(
        tok, emb, ph_emb, ph_tok, out, total_rows);
}